// SdcnTs_74148315398204
// MI455X (gfx1250) — compile-verified
//
#include <hip/hip_runtime.h>

typedef __attribute__((ext_vector_type(2))) float v2f;
typedef __attribute__((ext_vector_type(8))) float v8f;
typedef __attribute__((ext_vector_type(4))) int   v4i;

#define AS1 __attribute__((address_space(1)))
#define AS3 __attribute__((address_space(3)))

#if __has_builtin(__builtin_amdgcn_global_load_async_to_lds_b128) && \
    __has_builtin(__builtin_amdgcn_global_load_async_to_lds_b32)
#define HAS_ASYNC_LDS 1
#else
#define HAS_ASYNC_LDS 0
#endif

__device__ __forceinline__ void wait_async0() {
#if __has_builtin(__builtin_amdgcn_s_wait_asynccnt)
    __builtin_amdgcn_s_wait_asynccnt(0);
#else
    asm volatile("s_wait_asynccnt 0x0" ::: "memory");
#endif
}

namespace {
constexpr int BATCH = 2;
constexpr int NN    = 4096;
constexpr int F_IN  = 256;
constexpr int G     = 128;
constexpr int ZD    = 10;
constexpr int ZP    = 16;   // ZD padded to 16 for WMMA tiles
constexpr int KC    = 4;
constexpr float EPS_BN = 1e-5f;
}

// ---------------------------------------------------------------------------
// support = (ax*X + ay*Y) @ W   with optional Y (residual / z-broadcast),
// Y zero-extended to Fdim channels if yF < Fdim. Output padded to ldO cols.
// ---------------------------------------------------------------------------
__global__ __launch_bounds__(256)
void support_gemm(const float* __restrict__ X, long xBS, int ldX,
                  const float* __restrict__ Y, long yBS, int ldY, int yF,
                  const float* __restrict__ W,
                  float* __restrict__ out, int ldO,
                  int Fdim, int O, float ax, float ay) {
    long idx = (long)blockIdx.x * 256 + threadIdx.x;
    long total = (long)BATCH * NN * ldO;
    if (idx >= total) return;
    int o   = (int)(idx % ldO);
    long bn = idx / ldO;
    int b   = (int)(bn / NN);
    int n   = (int)(bn % NN);
    if (o >= O) { out[idx] = 0.0f; return; }  // keep pad columns zero
    const float* x = X + (long)b * xBS + (long)n * ldX;
    const float* y = Y ? (Y + (long)b * yBS + (long)n * ldY) : nullptr;
    float acc = 0.0f;
    for (int f = 0; f < Fdim; ++f) {
        float v = ax * x[f];
        if (y && f < yF) v += ay * y[f];
        acc += v * W[f * O + o];
    }
    out[idx] = acc;
}

// ---------------------------------------------------------------------------
// C[b] = adj[b] @ S[b] using V_WMMA_F32_16X16X4_F32.
// Block = 256 threads = 8 waves; block tile BM x BN; wave owns 16x16 subtiles.
// Double-buffered LDS, one barrier per K-step. Tiles are staged with
// GLOBAL_LOAD_ASYNC_TO_LDS_B128 (ASYNCcnt, no VGPR staging) when the builtins
// exist; otherwise the pipelined global->reg->ds_store path is used.
// A 16x4 frag: lane m=lane&15, kb=(lane>>4)*2 -> {A[m][k+kb], A[m][k+kb+1]}
// B 4x16 frag: lane n=lane&15            -> {B[k+kb][n], B[k+kb+1][n]}
// C/D: lanes 0-15 hold M=vgpr, lanes 16-31 hold M=vgpr+8, N=lane&15.
// ---------------------------------------------------------------------------
template <int BM, int BN>
__global__ __launch_bounds__(256)
void adj_gemm_wmma(const float* __restrict__ adj,
                   const float* __restrict__ S,
                   float* __restrict__ C, int ldB) {
    constexpr int BK   = 16;
    constexpr int SUB  = (BM / 16) * (BN / 16);
    constexpr int SPW  = SUB / 8;                   // subtiles per wave
    constexpr int APER = (BM * BK) / (256 * 4);     // float4 A-loads per thread
    constexpr bool BVEC = (BK * BN) >= 256 * 4;     // B tile loaded as float4?
    __shared__ __align__(16) float As[2][BM * BK];
    __shared__ __align__(16) float Bs[2][BK * BN];

    const int b    = blockIdx.z;
    const int row0 = blockIdx.x * BM;
    const int col0 = blockIdx.y * BN;
    const float* A  = adj + (long)b * NN * NN;
    const float* Bp = S   + (long)b * NN * ldB;
    float*       Cp = C   + (long)b * NN * ldB;

    const int tid  = threadIdx.x;
    const int wave = tid >> 5;
    const int lane = tid & 31;
    const int m    = lane & 15;
    const int kb   = (lane >> 4) << 1;

    // per-thread staging coordinates
    int ar[APER], ac[APER];
    for (int it = 0; it < APER; ++it) {
        int e = (tid + it * 256) * 4;
        ar[it] = e / BK; ac[it] = e % BK;
    }
    int br, bc;
    if constexpr (BVEC) { int e = tid * 4; br = e / BN; bc = e % BN; }
    else                {             br = tid / BN; bc = tid % BN; }

    v8f acc[SPW] = {};

    auto compute = [&](int cur) {
        for (int s = 0; s < SPW; ++s) {
            const int sub = wave + s * 8;
            const int sr  = sub / (BN / 16);
            const int sc  = sub % (BN / 16);
            const float* as = &As[cur][(sr * 16 + m) * BK];
            const float* bs = &Bs[cur][sc * 16 + m];
            for (int kk = 0; kk < BK; kk += 4) {
                v2f a, bf;
                a.x  = as[kk + kb];
                a.y  = as[kk + kb + 1];
                bf.x = bs[(kk + kb)     * BN];
                bf.y = bs[(kk + kb + 1) * BN];
                acc[s] = __builtin_amdgcn_wmma_f32_16x16x4_f32(
                    false, a, false, bf, (short)0, acc[s], false, false);
            }
        }
    };

#if HAS_ASYNC_LDS
    // ---------------- async global->LDS staging (ASYNCcnt) ----------------
    auto async_fill = [&](int k0, int buf) {
        for (int it = 0; it < APER; ++it)
            __builtin_amdgcn_global_load_async_to_lds_b128(
                (AS1 v4i*)&A[(long)(row0 + ar[it]) * NN + k0 + ac[it]],
                (AS3 v4i*)&As[buf][ar[it] * BK + ac[it]], 0, 0);
        if constexpr (BVEC)
            __builtin_amdgcn_global_load_async_to_lds_b128(
                (AS1 v4i*)&Bp[(long)(k0 + br) * ldB + col0 + bc],
                (AS3 v4i*)&Bs[buf][br * BN + bc], 0, 0);
        else if (tid < BK * BN)
            __builtin_amdgcn_global_load_async_to_lds_b32(
                (AS1 int*)&Bp[(long)(k0 + br) * ldB + col0 + bc],
                (AS3 int*)&Bs[buf][br * BN + bc], 0, 0);
    };

    async_fill(0, 0);
    wait_async0();
    __syncthreads();

    for (int k0 = 0; k0 < NN; k0 += BK) {
        const int  cur  = (k0 / BK) & 1;
        const bool more = (k0 + BK) < NN;
        if (more) async_fill(k0 + BK, cur ^ 1);         // DMA overlaps WMMAs
        if (k0 + 2 * BK < NN)                           // warm L2 two tiles out
            __builtin_prefetch(&A[(long)(row0 + ar[0]) * NN + k0 + 2 * BK], 0, 3);
        compute(cur);
        if (more) wait_async0();
        __syncthreads();
    }
#else
    // ------------- fallback: pipelined global->reg->LDS staging -------------
    float4 aReg[APER]; float4 bReg = {}; float bScal = 0.0f;
    auto gload = [&](int k0) {
        for (int it = 0; it < APER; ++it)
            aReg[it] = *reinterpret_cast<const float4*>(
                &A[(long)(row0 + ar[it]) * NN + k0 + ac[it]]);
        if constexpr (BVEC)
            bReg = *reinterpret_cast<const float4*>(
                &Bp[(long)(k0 + br) * ldB + col0 + bc]);
        else if (tid < BK * BN)
            bScal = Bp[(long)(k0 + br) * ldB + col0 + bc];
    };
    auto sstore = [&](int buf) {
        for (int it = 0; it < APER; ++it)
            *reinterpret_cast<float4*>(&As[buf][ar[it] * BK + ac[it]]) = aReg[it];
        if constexpr (BVEC)
            *reinterpret_cast<float4*>(&Bs[buf][br * BN + bc]) = bReg;
        else if (tid < BK * BN)
            Bs[buf][br * BN + bc] = bScal;
    };

    gload(0);
    sstore(0);
    for (int k0 = 0; k0 < NN; k0 += BK) {
        const int  cur  = (k0 / BK) & 1;
        const bool more = (k0 + BK) < NN;
        __syncthreads();
        if (more) gload(k0 + BK);
        if (k0 + 2 * BK < NN)
            __builtin_prefetch(&A[(long)(row0 + ar[0]) * NN + k0 + 2 * BK], 0, 3);
        compute(cur);
        if (more) sstore(cur ^ 1);
    }
#endif

    const int rbase = (lane >> 4) * 8;
    for (int s = 0; s < SPW; ++s) {
        const int sub = wave + s * 8;
        const int sr  = sub / (BN / 16);
        const int sc  = sub % (BN / 16);
        for (int vg = 0; vg < 8; ++vg) {
            int r = row0 + sr * 16 + rbase + vg;
            int c = col0 + sc * 16 + m;
            Cp[(long)r * ldB + c] = acc[s][vg];
        }
    }
}

// ---------------------------------------------------------------------------
// BatchNorm training-mode stats per channel over (B, N) rows.
// stats[2c] = mean, stats[2c+1] = rsqrt(var + eps)
// ---------------------------------------------------------------------------
__global__ __launch_bounds__(256)
void bn_stats(const float* __restrict__ X, int ldX, int O, float* __restrict__ stats) {
    const int c = blockIdx.x;
    if (c >= O) return;
    const long rows = (long)BATCH * NN;
    float s = 0.0f, s2 = 0.0f;
    for (long i = threadIdx.x; i < rows; i += 256) {
        float v = X[i * ldX + c];
        s += v; s2 += v * v;
    }
    __shared__ float rs[256], rq[256];
    rs[threadIdx.x] = s; rq[threadIdx.x] = s2;
    __syncthreads();
    for (int off = 128; off > 0; off >>= 1) {
        if (threadIdx.x < off) {
            rs[threadIdx.x] += rs[threadIdx.x + off];
            rq[threadIdx.x] += rq[threadIdx.x + off];
        }
        __syncthreads();
    }
    if (threadIdx.x == 0) {
        float inv = 1.0f / (float)rows;
        float mean = rs[0] * inv;
        float var  = rq[0] * inv - mean * mean;
        stats[2 * c]     = mean;
        stats[2 * c + 1] = rsqrtf(var + EPS_BN);
    }
}

__global__ __launch_bounds__(256)
void bn_apply(float* __restrict__ X, int ldX, int O,
              const float* __restrict__ stats,
              const float* __restrict__ gamma, const float* __restrict__ beta,
              int relu) {
    long idx = (long)blockIdx.x * 256 + threadIdx.x;
    long total = (long)BATCH * NN * ldX;
    if (idx >= total) return;
    int c = (int)(idx % ldX);
    if (c >= O) { X[idx] = 0.0f; return; }
    float v = (X[idx] - stats[2 * c]) * stats[2 * c + 1] * gamma[c] + beta[c];
    if (relu) v = fmaxf(v, 0.0f);
    X[idx] = v;
}

// ---------------------------------------------------------------------------
// student-t cluster distribution q [N, K]
// ---------------------------------------------------------------------------
__global__ __launch_bounds__(256)
void q_kernel(const float* __restrict__ z, const float* __restrict__ cluster,
              float* __restrict__ qout) {
    int n = blockIdx.x * 256 + threadIdx.x;
    if (n >= NN) return;
    float qv[KC]; float s = 0.0f;
    for (int k = 0; k < KC; ++k) {
        float d2 = 0.0f;
        for (int j = 0; j < ZD; ++j) {
            float d = z[n * ZD + j] - cluster[k * ZD + j];
            d2 += d * d;
        }
        qv[k] = 1.0f / (1.0f + d2);
        s += qv[k];
    }
    float inv = 1.0f / s;
    for (int k = 0; k < KC; ++k) qout[n * KC + k] = qv[k] * inv;
}

__global__ __launch_bounds__(256)
void pack_h5(const float* __restrict__ h5p, float* __restrict__ out) {
    long idx = (long)blockIdx.x * 256 + threadIdx.x;
    if (idx >= (long)BATCH * NN * ZD) return;
    int j   = (int)(idx % ZD);
    long bn = idx / ZD;
    out[idx] = h5p[bn * ZP + j];
}

// ---------------------------------------------------------------------------
// pred = softmax over node axis of z-score(h5) per (batch, channel)
// ---------------------------------------------------------------------------
__global__ __launch_bounds__(256)
void zsoftmax(const float* __restrict__ h5p, float* __restrict__ pred) {
    const int b = blockIdx.x / ZD;
    const int j = blockIdx.x % ZD;
    const float* x = h5p + (long)b * NN * ZP + j;
    const int tid = threadIdx.x;
    __shared__ float red[256];
    __shared__ float sh_mu, sh_rsd, sh_max, sh_sum;

    float s = 0.0f, s2 = 0.0f, mx = -3.0e38f;
    for (int n = tid; n < NN; n += 256) {
        float v = x[(long)n * ZP];
        s += v; s2 += v * v; mx = fmaxf(mx, v);
    }
    red[tid] = s; __syncthreads();
    for (int off = 128; off > 0; off >>= 1) {
        if (tid < off) red[tid] += red[tid + off];
        __syncthreads();
    }
    if (tid == 0) sh_mu = red[0] / (float)NN;
    __syncthreads();
    red[tid] = s2; __syncthreads();
    for (int off = 128; off > 0; off >>= 1) {
        if (tid < off) red[tid] += red[tid + off];
        __syncthreads();
    }
    if (tid == 0) {
        float var = red[0] / (float)NN - sh_mu * sh_mu;
        sh_rsd = rsqrtf(fmaxf(var, 1.0e-30f));
    }
    __syncthreads();
    red[tid] = mx; __syncthreads();
    for (int off = 128; off > 0; off >>= 1) {
        if (tid < off) red[tid] = fmaxf(red[tid], red[tid + off]);
        __syncthreads();
    }
    if (tid == 0) sh_max = (red[0] - sh_mu) * sh_rsd;
    __syncthreads();
    const float mu = sh_mu, rsd = sh_rsd, ym = sh_max;
    float es = 0.0f;
    for (int n = tid; n < NN; n += 256) {
        float v = (x[(long)n * ZP] - mu) * rsd;
        es += __expf(v - ym);
    }
    red[tid] = es; __syncthreads();
    for (int off = 128; off > 0; off >>= 1) {
        if (tid < off) red[tid] += red[tid + off];
        __syncthreads();
    }
    if (tid == 0) sh_sum = red[0];
    __syncthreads();
    const float inv = 1.0f / sh_sum;
    for (int n = tid; n < NN; n += 256) {
        float v = (x[(long)n * ZP] - mu) * rsd;
        pred[((long)b * NN + n) * ZD + j] = __expf(v - ym) * inv;
    }
}

// ---------------------------------------------------------------------------
extern "C" void kernel_launch(void* const* d_in, const int* in_sizes, int n_in,
                              void* d_out, int out_size, void* d_ws, size_t ws_size,
                              hipStream_t stream) {
    const float* enc  = (const float*)d_in[0];
    const float* tra1 = (const float*)d_in[1];
    const float* tra2 = (const float*)d_in[2];
    const float* tra3 = (const float*)d_in[3];
    const float* z    = (const float*)d_in[4];
    const float* adj  = (const float*)d_in[5];
    const float* W1 = (const float*)d_in[6],  *g1 = (const float*)d_in[7],  *b1 = (const float*)d_in[8];
    const float* W2 = (const float*)d_in[9],  *g2 = (const float*)d_in[10], *b2 = (const float*)d_in[11];
    const float* W3 = (const float*)d_in[12], *g3 = (const float*)d_in[13], *b3 = (const float*)d_in[14];
    const float* W4 = (const float*)d_in[15], *g4 = (const float*)d_in[16], *b4 = (const float*)d_in[17];
    const float* W5 = (const float*)d_in[18], *g5 = (const float*)d_in[19], *b5 = (const float*)d_in[20];
    const float* cluster = (const float*)d_in[21];

    const long BN_ = (long)BATCH * NN;
    float* ws   = (float*)d_ws;
    float* sup  = ws;                 // B*N*128
    float* hA   = sup + BN_ * G;      // B*N*128
    float* hB   = hA  + BN_ * G;      // B*N*128
    float* h4p  = hB  + BN_ * G;      // B*N*16
    float* h5p  = h4p + BN_ * ZP;     // B*N*16
    float* stats = h5p + BN_ * ZP;    // 2*128

    float* q_out    = (float*)d_out;
    float* pred_out = q_out + (long)NN * KC;
    float* h5_out   = pred_out + BN_ * ZD;

    const int blkBig = (int)((BN_ * G  + 255) / 256);   // 4096
    const int blkPad = (int)((BN_ * ZP + 255) / 256);   // 512
    const dim3 gridBig(NN / 64, G / 64, BATCH);
    const dim3 gridSm (NN / 128, 1, BATCH);
    const float SG = 0.5f;

    // ---- layer 1: h1 = BN(adj @ (enc @ W1)), relu ----
    support_gemm<<<blkBig, 256, 0, stream>>>(enc, (long)NN * F_IN, F_IN,
                                             nullptr, 0, 0, 0,
                                             W1, sup, G, F_IN, G, 1.0f, 0.0f);
    adj_gemm_wmma<64, 64><<<gridBig, 256, 0, stream>>>(adj, sup, hA, G);
    bn_stats<<<G, 256, 0, stream>>>(hA, G, G, stats);
    bn_apply<<<blkBig, 256, 0, stream>>>(hA, G, G, stats, g1, b1, 1);

    // ---- layer 2 ----
    support_gemm<<<blkBig, 256, 0, stream>>>(hA, (long)NN * G, G,
                                             tra1, (long)NN * G, G, G,
                                             W2, sup, G, G, G, 1.0f - SG, SG);
    adj_gemm_wmma<64, 64><<<gridBig, 256, 0, stream>>>(adj, sup, hB, G);
    bn_stats<<<G, 256, 0, stream>>>(hB, G, G, stats);
    bn_apply<<<blkBig, 256, 0, stream>>>(hB, G, G, stats, g2, b2, 1);

    // ---- layer 3 ----
    support_gemm<<<blkBig, 256, 0, stream>>>(hB, (long)NN * G, G,
                                             tra2, (long)NN * G, G, G,
                                             W3, sup, G, G, G, 1.0f - SG, SG);
    adj_gemm_wmma<64, 64><<<gridBig, 256, 0, stream>>>(adj, sup, hA, G);
    bn_stats<<<G, 256, 0, stream>>>(hA, G, G, stats);
    bn_apply<<<blkBig, 256, 0, stream>>>(hA, G, G, stats, g3, b3, 1);

    // ---- layer 4 (tra3 has ZD channels; zero-extended to G) ----
    support_gemm<<<blkPad, 256, 0, stream>>>(hA, (long)NN * G, G,
                                             tra3, (long)NN * ZD, ZD, ZD,
                                             W4, sup, ZP, G, ZD, 1.0f - SG, SG);
    adj_gemm_wmma<128, 16><<<gridSm, 256, 0, stream>>>(adj, sup, h4p, ZP);
    bn_stats<<<ZD, 256, 0, stream>>>(h4p, ZP, ZD, stats);
    bn_apply<<<blkPad, 256, 0, stream>>>(h4p, ZP, ZD, stats, g4, b4, 1);

    // ---- layer 5 (z broadcast over batch; no relu) ----
    support_gemm<<<blkPad, 256, 0, stream>>>(h4p, (long)NN * ZP, ZP,
                                             z, 0, ZD, ZD,
                                             W5, sup, ZP, ZD, ZD, 1.0f - SG, SG);
    adj_gemm_wmma<128, 16><<<gridSm, 256, 0, stream>>>(adj, sup, h5p, ZP);
    bn_stats<<<ZD, 256, 0, stream>>>(h5p, ZP, ZD, stats);
    bn_apply<<<blkPad, 256, 0, stream>>>(h5p, ZP, ZD, stats, g5, b5, 0);

    // ---- outputs ----
    q_kernel<<<(NN + 255) / 256, 256, 0, stream>>>(z, cluster, q_out);
    pack_h5<<<(int)((BN_ * ZD + 255) / 256), 256, 0, stream>>>(h5p, h5_out);
    zsoftmax<<<BATCH * ZD, 256, 0, stream>>>(h5p, pred_out);
}